// GraphClassificationNetwork_31061203484851
// MI455X (gfx1250) — compile-verified
//
#include <hip/hip_runtime.h>
#include <math.h>

// MI455X (gfx1250) implementation notes:
//  - wave32 only; all matrix math via V_WMMA_F32_16X16X4_F32 (fp32-faithful,
//    D = A(16x4) x B(4x16) + C(16x16), chained over K in steps of 4).
//  - Edge MLPs: weights staged in LDS (<=64KB), 16 edges/wave, 4 N-tiles of
//    16 cols kept in 4 v8f accumulators so each gathered A-frag feeds 4 WMMAs.
//  - Scatters use global_atomic_add_f32 (unsafeAtomicAdd); reductions use
//    block partials + few global atomics (memory-bound problem, 23.3 TB/s HBM).

typedef float v2f __attribute__((ext_vector_type(2)));
typedef float v8f __attribute__((ext_vector_type(8)));

#define NNODES 50000
#define NEDGES 800000

// ---------------------------------------------------------------------------
// Generic fp32 WMMA GEMM: C[M,Nout] = act(A[M,K] @ W[K,Nout] + bias + colvec)
// One wave (32 threads) computes one 16x16 tile.
// Fragment layouts per CDNA5 ISA 7.12.2:
//   A 16x4 : lanes 0-15 row=lane K={0,1}; lanes 16-31 row=lane-16 K={2,3}
//   B 4x16 : VGPR0 lanes0-15 K=0, lanes16-31 K=2 ; VGPR1 K=1 / K=3
//   C 16x16: VGPR r -> row m0+r (lanes<16) / m0+8+r (lanes>=16), col n0+lane%16
// ---------------------------------------------------------------------------
__global__ void k_gemm(const float* __restrict__ A, const float* __restrict__ W,
                       const float* __restrict__ bias, const float* __restrict__ colvec,
                       float* __restrict__ C, int M, int K, int Nout, int relu_flag)
{
    int lane = threadIdx.x & 31;
    int half = lane >> 4;
    int mr   = lane & 15;
    int m0   = blockIdx.x << 4;
    int n0   = blockIdx.y << 4;

    int arow = m0 + mr; if (arow > M - 1) arow = M - 1;   // clamp pad rows
    const float* ap = A + (size_t)arow * K + (half << 1);

    v8f acc = {};
    for (int k = 0; k < K; k += 4) {
        v2f a; a.x = ap[k]; a.y = ap[k + 1];
        const float* wp = W + (size_t)(k + (half << 1)) * Nout + n0 + mr;
        v2f b; b.x = wp[0]; b.y = wp[Nout];
        acc = __builtin_amdgcn_wmma_f32_16x16x4_f32(false, a, false, b,
                                                    (short)0, acc, false, false);
    }

    float badd = (bias ? bias[n0 + mr] : 0.0f) + (colvec ? colvec[n0 + mr] : 0.0f);
#pragma unroll
    for (int r = 0; r < 8; r++) {
        int m = m0 + half * 8 + r;
        if (m < M) {
            float v = acc[r] + badd;
            if (relu_flag) v = fmaxf(v, 0.0f);
            C[(size_t)m * Nout + n0 + mr] = v;
        }
    }
}

// ---------------------------------------------------------------------------
// Gated-layer edge GEMM:
//   eout[i,:] = h[src]@Aw + h[dst]@Bw + e[dst]@Cw + e[src]@Dw + biassum
// Weights staged in LDS; 8 waves/block, each wave = 16 edges x 64 cols.
// ---------------------------------------------------------------------------
template<int DINH>
__global__ void k_gated_eout(const float* __restrict__ h, const float* __restrict__ e,
                             const int* __restrict__ src, const int* __restrict__ dst,
                             const float* __restrict__ Aw, const float* __restrict__ Bw,
                             const float* __restrict__ Cw, const float* __restrict__ Dw,
                             const float* __restrict__ bsum,
                             float* __restrict__ eout, int E_)
{
    __shared__ float sW[2 * DINH * 64 + 2 * 64 * 64];
    float* sA = sW;
    float* sB = sA + DINH * 64;
    float* sC = sB + DINH * 64;
    float* sD = sC + 64 * 64;
    for (int i = threadIdx.x; i < DINH * 64; i += 256) { sA[i] = Aw[i]; sB[i] = Bw[i]; }
    for (int i = threadIdx.x; i < 64 * 64;  i += 256) { sC[i] = Cw[i]; sD[i] = Dw[i]; }
    __syncthreads();

    int wid  = threadIdx.x >> 5;
    int lane = threadIdx.x & 31;
    int half = lane >> 4;
    int mr   = lane & 15;
    int eb   = (blockIdx.x * 8 + wid) * 16;

    int eidx = eb + mr; if (eidx > E_ - 1) eidx = E_ - 1;
    int rs = src[eidx], rd = dst[eidx];
    const float* gA = h + (size_t)rs * DINH;   // h[src]
    const float* gB = h + (size_t)rd * DINH;   // h[dst]
    const float* gC = e + (size_t)rd * 64;     // e[dst] (node ids index e; faithful)
    const float* gD = e + (size_t)rs * 64;     // e[src]

    v8f acc[4] = {};
    auto term = [&](const float* g, const float* lw, int K) {
        for (int k = 0; k < K; k += 4) {
            v2f a; a.x = g[k + half * 2]; a.y = g[k + half * 2 + 1];
#pragma unroll
            for (int nt = 0; nt < 4; nt++) {
                const float* wp = lw + (k + half * 2) * 64 + nt * 16 + mr;
                v2f b; b.x = wp[0]; b.y = wp[64];
                acc[nt] = __builtin_amdgcn_wmma_f32_16x16x4_f32(false, a, false, b,
                                                                (short)0, acc[nt],
                                                                false, false);
            }
        }
    };
    term(gA, sA, DINH);
    term(gB, sB, DINH);
    term(gC, sC, 64);
    term(gD, sD, 64);

#pragma unroll
    for (int nt = 0; nt < 4; nt++) {
        int col = nt * 16 + mr;
        float badd = bsum[col];
#pragma unroll
        for (int r = 0; r < 8; r++) {
            int row = eb + half * 8 + r;
            if (row < E_) eout[(size_t)row * 64 + col] = acc[nt][r] + badd;
        }
    }
}

// ---------------------------------------------------------------------------
// Utility kernels
// ---------------------------------------------------------------------------
__global__ void k_zero(float* p, int n)
{
    int i = blockIdx.x * blockDim.x + threadIdx.x;
    if (i < n) p[i] = 0.0f;
}

// weighted edge scatter: agg[dst] += w * y[src]  (agg pre-initialized to y)
__global__ void k_edge_scatter(const float* __restrict__ y, float* __restrict__ agg,
                               const int* __restrict__ src, const int* __restrict__ dst,
                               const float* __restrict__ w, int E_, int Fq, int F)
{
    int idx = blockIdx.x * blockDim.x + threadIdx.x;
    if (idx >= E_ * Fq) return;
    int i = idx / Fq, j = idx % Fq;
    float wv = w[i];
    float4 v = ((const float4*)(y + (size_t)src[i] * F))[j];
    float* ap = agg + (size_t)dst[i] * F + j * 4;
    unsafeAtomicAdd(ap + 0, wv * v.x);
    unsafeAtomicAdd(ap + 1, wv * v.y);
    unsafeAtomicAdd(ap + 2, wv * v.z);
    unsafeAtomicAdd(ap + 3, wv * v.w);
}

// per-node relu + L2 normalize (one wave per node)
__global__ void k_relu_norm(const float* __restrict__ in, float* __restrict__ out,
                            int n, int F)
{
    int wid = threadIdx.x >> 5, lane = threadIdx.x & 31;
    int node = blockIdx.x * 8 + wid;
    if (node >= n) return;                      // uniform per-wave
    int per = F >> 5;
    float r[4]; float ss = 0.0f;
    const float* ip = in + (size_t)node * F;
    for (int j = 0; j < per; j++) { float v = fmaxf(ip[lane + j * 32], 0.0f); r[j] = v; ss += v * v; }
    for (int m = 16; m > 0; m >>= 1) ss += __shfl_xor(ss, m, 32);
    float inv = 1.0f / fmaxf(sqrtf(ss), 1e-12f);
    float* op = out + (size_t)node * F;
    for (int j = 0; j < per; j++) op[lane + j * 32] = r[j] * inv;
}

// e[i] = concat(h[src[i]], h[dst[i]])   (h rows are 32 floats)
__global__ void k_build_e(const float* __restrict__ h, const int* __restrict__ src,
                          const int* __restrict__ dst, float* __restrict__ e, int E_)
{
    int idx = blockIdx.x * blockDim.x + threadIdx.x;
    if (idx >= E_ * 16) return;
    int i = idx >> 4, j = idx & 15;
    int node = (j < 8) ? src[i] : dst[i];
    ((float4*)(e + (size_t)i * 64))[j] =
        ((const float4*)(h + (size_t)node * 32))[j & 7];
}

// per-column sum / sumsq over rows (cols fixed = 64)
__global__ void k_colstats(const float* __restrict__ X, int rows,
                           float* __restrict__ sum, float* __restrict__ sumsq)
{
    __shared__ float s1[256];
    __shared__ float s2[256];
    int t = threadIdx.x, c = t & 63, rg = t >> 6;
    float a0 = 0.0f, a1 = 0.0f;
    for (int r = blockIdx.x * 4 + rg; r < rows; r += gridDim.x * 4) {
        float v = X[(size_t)r * 64 + c];
        a0 += v; a1 += v * v;
    }
    s1[t] = a0; s2[t] = a1;
    __syncthreads();
    if (t < 64) {
        unsafeAtomicAdd(&sum[t], s1[t] + s1[t + 64] + s1[t + 128] + s1[t + 192]);
        if (sumsq) unsafeAtomicAdd(&sumsq[t], s2[t] + s2[t + 64] + s2[t + 128] + s2[t + 192]);
    }
}

__global__ void k_bias4(const float* a, const float* b, const float* c, const float* d,
                        float* out)
{
    int t = threadIdx.x;
    out[t] = a[t] + b[t] + c[t] + d[t];
}

__global__ void k_finalize_bn(const float* sum, const float* sumsq,
                              const float* g, const float* b,
                              float* scale, float* shift, int rows)
{
    int c = threadIdx.x;
    float mean = sum[c] / (float)rows;
    float var  = sumsq[c] / (float)rows - mean * mean;
    float sc   = g[c] * rsqrtf(var + 1e-5f);
    scale[c] = sc;
    shift[c] = b[c] - mean * sc;
}

// e[src[i]] += relu(scale*eout[i] + shift)
__global__ void k_bn_relu_scatter(const float* __restrict__ eout,
                                  const float* __restrict__ scale,
                                  const float* __restrict__ shift,
                                  const int* __restrict__ src,
                                  float* __restrict__ e, int E_)
{
    int idx = blockIdx.x * blockDim.x + threadIdx.x;
    if (idx >= E_ * 64) return;
    int i = idx >> 6, c = idx & 63;
    float v = fmaxf(scale[c] * eout[idx] + shift[c], 0.0f);
    unsafeAtomicAdd(&e[(size_t)src[i] * 64 + c], v);
}

__global__ void k_sigmoid(float* p, int n)
{
    int i = blockIdx.x * blockDim.x + threadIdx.x;
    if (i < n) p[i] = 1.0f / (1.0f + __expf(-p[i]));
}

// e := e/(colsum+1e-5); svec += sum_i e_new[i]*hv[dst[i]]
__global__ void k_colscale_svec(float* __restrict__ e, const float* __restrict__ colsum,
                                const float* __restrict__ hv, const int* __restrict__ dst,
                                float* __restrict__ svec, int rows)
{
    __shared__ float s1[256];
    int t = threadIdx.x, c = t & 63, rg = t >> 6;
    float inv = 1.0f / (colsum[c] + 1e-5f);
    float acc = 0.0f;
    for (int r = blockIdx.x * 4 + rg; r < rows; r += gridDim.x * 4) {
        size_t idx = (size_t)r * 64 + c;
        float en = e[idx] * inv;
        e[idx] = en;
        acc += en * hv[(size_t)dst[r] * 64 + c];
    }
    s1[t] = acc;
    __syncthreads();
    if (t < 64) unsafeAtomicAdd(&svec[t], s1[t] + s1[t + 64] + s1[t + 128] + s1[t + 192]);
}

// g = hsum/N ; logits = g@clsW + clsb ; log_softmax -> out[3]
__global__ void k_final_head(const float* __restrict__ hsum, const float* __restrict__ clsW,
                             const float* __restrict__ clsb, float* __restrict__ out, int n)
{
    __shared__ float g[64];
    __shared__ float lg[3];
    int t = threadIdx.x;
    g[t] = hsum[t] / (float)n;
    __syncthreads();
    if (t < 3) {
        float s = clsb[t];
        for (int c = 0; c < 64; c++) s += g[c] * clsW[c * 3 + t];
        lg[t] = s;
    }
    __syncthreads();
    if (t == 0) {
        float m = fmaxf(lg[0], fmaxf(lg[1], lg[2]));
        float se = __expf(lg[0] - m) + __expf(lg[1] - m) + __expf(lg[2] - m);
        float ls = m + __logf(se);
        out[0] = lg[0] - ls; out[1] = lg[1] - ls; out[2] = lg[2] - ls;
    }
}

// ---------------------------------------------------------------------------
extern "C" void kernel_launch(void* const* d_in, const int* in_sizes, int n_in,
                              void* d_out, int out_size, void* d_ws, size_t ws_size,
                              hipStream_t stream)
{
    (void)in_sizes; (void)n_in; (void)out_size; (void)ws_size;
    const int N = NNODES, E = NEDGES;

    const float* x  = (const float*)d_in[0];
    const int*   ei = (const int*)d_in[1];
    const float* ew = (const float*)d_in[2];
    const int* src = ei;
    const int* dst = ei + E;

    int pi = 3;
    auto P = [&]() -> const float* { return (const float*)d_in[pi++]; };
    const float *gcn1_W = P(), *gcn1_b = P(), *gcn2_W = P(), *gcn2_b = P();
    struct GP { const float *Aw,*Ab,*Bw,*Bb,*Cw,*Cb,*Dw,*Db,*bng,*bnb,*Uw,*Ub,*Vw,*Vb; };
    GP g1, g2;
    g1.Aw=P(); g1.Ab=P(); g1.Bw=P(); g1.Bb=P(); g1.Cw=P(); g1.Cb=P(); g1.Dw=P(); g1.Db=P();
    g1.bng=P(); g1.bnb=P(); g1.Uw=P(); g1.Ub=P(); g1.Vw=P(); g1.Vb=P();
    g2.Aw=P(); g2.Ab=P(); g2.Bw=P(); g2.Bb=P(); g2.Cw=P(); g2.Cb=P(); g2.Dw=P(); g2.Db=P();
    g2.bng=P(); g2.bnb=P(); g2.Uw=P(); g2.Ub=P(); g2.Vw=P(); g2.Vb=P();
    const float *clsW = P(), *clsb = P();

    // workspace layout (floats)
    float* pool = (float*)d_ws;
    float* e   = pool;                      // E*64  edge features (persistent per call)
    float* R1  = e  + (size_t)E * 64;       // E*64  eout / y1 / agg / agg2 / hv (reused)
    float* hA  = R1 + (size_t)E * 64;       // N*64
    float* hB  = hA + (size_t)N * 64;       // N*64
    float* st  = hB + (size_t)N * 64;       // 512 stats
    float *bnsum = st, *bnsumsq = st + 64, *scale = st + 128, *shift = st + 192,
          *colsum = st + 256, *svec = st + 320, *hsum = st + 384, *bias4 = st + 448;

    float* y1   = R1;                        // N*128
    float* agg  = R1 + (size_t)N * 128;      // N*128
    float* agg2 = R1 + (size_t)N * 256;      // N*32

    const int th = 256;
    dim3 wave(32);

    // ---- GCN layer 1 (128 -> 128) ----
    k_gemm<<<dim3((N + 15) / 16, 8), wave, 0, stream>>>(x, gcn1_W, gcn1_b, nullptr,
                                                        y1, N, 128, 128, 0);
    hipMemcpyAsync(agg, y1, (size_t)N * 128 * sizeof(float),
                   hipMemcpyDeviceToDevice, stream);
    k_edge_scatter<<<(E * 32 + th - 1) / th, th, 0, stream>>>(y1, agg, src, dst, ew, E, 32, 128);
    k_relu_norm<<<(N + 7) / 8, 256, 0, stream>>>(agg, agg, N, 128);

    // ---- GCN layer 2 (128 -> 32) ----
    k_gemm<<<dim3((N + 15) / 16, 2), wave, 0, stream>>>(agg, gcn2_W, gcn2_b, nullptr,
                                                        y1, N, 128, 32, 0);
    hipMemcpyAsync(agg2, y1, (size_t)N * 32 * sizeof(float),
                   hipMemcpyDeviceToDevice, stream);
    k_edge_scatter<<<(E * 8 + th - 1) / th, th, 0, stream>>>(y1, agg2, src, dst, ew, E, 8, 32);
    k_relu_norm<<<(N + 7) / 8, 256, 0, stream>>>(agg2, hA, N, 32);

    // ---- edge features e = [h[src], h[dst]] ----
    k_build_e<<<(E * 16 + th - 1) / th, th, 0, stream>>>(hA, src, dst, e, E);

    // ---- gated layers ----
    auto gated = [&](int dinh, const float* hin, float* hout, const GP& g) {
        float* eout = R1;
        float* hv   = R1;   // reuses eout region after it is consumed
        k_zero<<<2, 256, 0, stream>>>(st, 384);
        k_bias4<<<1, 64, 0, stream>>>(g.Ab, g.Bb, g.Cb, g.Db, bias4);
        if (dinh == 32)
            k_gated_eout<32><<<(E + 127) / 128, 256, 0, stream>>>(hin, e, src, dst,
                g.Aw, g.Bw, g.Cw, g.Dw, bias4, eout, E);
        else
            k_gated_eout<64><<<(E + 127) / 128, 256, 0, stream>>>(hin, e, src, dst,
                g.Aw, g.Bw, g.Cw, g.Dw, bias4, eout, E);
        k_colstats<<<1024, 256, 0, stream>>>(eout, E, bnsum, bnsumsq);
        k_finalize_bn<<<1, 64, 0, stream>>>(bnsum, bnsumsq, g.bng, g.bnb, scale, shift, E);
        k_bn_relu_scatter<<<(E * 64 + th - 1) / th, th, 0, stream>>>(eout, scale, shift, src, e, E);
        k_sigmoid<<<(E * 64 + th - 1) / th, th, 0, stream>>>(e, E * 64);
        k_colstats<<<1024, 256, 0, stream>>>(e, E, colsum, nullptr);
        k_gemm<<<dim3((N + 15) / 16, 4), wave, 0, stream>>>(hin, g.Vw, g.Vb, nullptr,
                                                            hv, N, dinh, 64, 0);
        k_colscale_svec<<<1024, 256, 0, stream>>>(e, colsum, hv, dst, svec, E);
        k_gemm<<<dim3((N + 15) / 16, 4), wave, 0, stream>>>(hin, g.Uw, g.Ub, svec,
                                                            hout, N, dinh, 64, 1);
    };
    gated(32, hA, hB, g1);
    gated(64, hB, hA, g2);

    // ---- head: graph mean -> classifier -> log_softmax ----
    k_zero<<<1, 64, 0, stream>>>(hsum, 64);
    k_colstats<<<512, 256, 0, stream>>>(hA, N, hsum, nullptr);
    k_final_head<<<1, 64, 0, stream>>>(hsum, clsW, clsb, (float*)d_out, N);
}